// LocalGrouper_81836306858509
// MI455X (gfx1250) — compile-verified
//
#include <hip/hip_runtime.h>
#include <hip/hip_bf16.h>
#include <math.h>

// Problem constants (from reference setup_inputs)
#define BB 4
#define NN 4096
#define DD 256
#define SS (NN / 2)       // 2048
#define KK 24
#define CIN (DD + 3)      // 259
#define COUT (2 * DD + 3) // 515
#define EPS 1e-5f

typedef float v2f __attribute__((ext_vector_type(2)));
typedef float v8f __attribute__((ext_vector_type(8)));

// ---------------------------------------------------------------------------
// Kernel 1: furthest point sampling (1 block per batch) + bitonic sort of picks
// ---------------------------------------------------------------------------
__global__ void fps_kernel(const float* __restrict__ xyz, int* __restrict__ fps_idx) {
  const int b = blockIdx.x;
  const int tid = threadIdx.x;               // 256 threads = 8 waves (wave32)
  const int lane = tid & 31;
  const int wave = tid >> 5;
  const float* X = xyz + (size_t)b * NN * 3;

  __shared__ float s_dist[NN];               // 16 KB
  __shared__ int   s_pick[SS];               // 8 KB
  __shared__ float s_last[3];
  __shared__ float s_wv[8];
  __shared__ int   s_wi[8];

  for (int i = tid; i < NN; i += 256) s_dist[i] = 1e10f;
  if (tid == 0) {
    s_pick[0] = 0;
    s_last[0] = X[0]; s_last[1] = X[1]; s_last[2] = X[2];
  }
  __syncthreads();

  for (int it = 1; it < SS; ++it) {
    const float lx = s_last[0], ly = s_last[1], lz = s_last[2];
    float bestv = -1.0f; int besti = 0;
    for (int i = tid; i < NN; i += 256) {
      float dx = X[3 * i + 0] - lx;
      float dy = X[3 * i + 1] - ly;
      float dz = X[3 * i + 2] - lz;
      float d = dx * dx + dy * dy + dz * dz;
      float cur = fminf(s_dist[i], d);
      s_dist[i] = cur;
      if (cur > bestv) { bestv = cur; besti = i; }
    }
    // wave32 shuffle argmax (first-index tiebreak like jnp.argmax)
    for (int off = 16; off > 0; off >>= 1) {
      float ov = __shfl_xor(bestv, off, 32);
      int   oi = __shfl_xor(besti, off, 32);
      if (ov > bestv || (ov == bestv && oi < besti)) { bestv = ov; besti = oi; }
    }
    if (lane == 0) { s_wv[wave] = bestv; s_wi[wave] = besti; }
    __syncthreads();
    if (tid == 0) {
      float bv = s_wv[0]; int bi = s_wi[0];
      for (int w = 1; w < 8; ++w)
        if (s_wv[w] > bv || (s_wv[w] == bv && s_wi[w] < bi)) { bv = s_wv[w]; bi = s_wi[w]; }
      s_pick[it] = bi;
      s_last[0] = X[3 * bi + 0]; s_last[1] = X[3 * bi + 1]; s_last[2] = X[3 * bi + 2];
    }
    __syncthreads();
  }

  // bitonic sort ascending (SS = 2048 is a power of two)
  for (int k = 2; k <= SS; k <<= 1) {
    for (int j = k >> 1; j > 0; j >>= 1) {
      for (int t = tid; t < SS; t += 256) {
        int ixj = t ^ j;
        if (ixj > t) {
          int a = s_pick[t], c = s_pick[ixj];
          bool up = ((t & k) == 0);
          if ((a > c) == up) { s_pick[t] = c; s_pick[ixj] = a; }
        }
      }
      __syncthreads();
    }
  }
  for (int t = tid; t < SS; t += 256) fps_idx[b * SS + t] = s_pick[t];
}

// ---------------------------------------------------------------------------
// Kernel 2: gather new_xyz (-> out + ws) and new_points_res (-> out)
// ---------------------------------------------------------------------------
__global__ void gather_kernel(const float* __restrict__ xyz,
                              const float* __restrict__ points_res,
                              const int* __restrict__ fps_idx,
                              float* __restrict__ out_xyz,
                              float* __restrict__ new_xyz_ws,
                              float* __restrict__ out_res) {
  const int bs = blockIdx.x;           // 0 .. B*S-1
  const int b = bs / SS;
  const int id = fps_idx[bs];
  const size_t src = ((size_t)b * NN + id);
  for (int c = threadIdx.x; c < DD; c += blockDim.x)
    out_res[(size_t)bs * DD + c] = points_res[src * DD + c];
  if (threadIdx.x < 3) {
    float v = xyz[src * 3 + threadIdx.x];
    out_xyz[(size_t)bs * 3 + threadIdx.x] = v;
    new_xyz_ws[(size_t)bs * 3 + threadIdx.x] = v;
  }
}

// ---------------------------------------------------------------------------
// Kernel 3: KNN via V_WMMA_F32_16X16X4_F32 distance tiles + top-24 select
//   dist(q,p) ~ -2*q.p + |p|^2   (row-constant |q|^2 dropped; ordering equal)
//   A row m = [-2qx, -2qy, -2qz, 1]; B col n = [px, py, pz, |p|^2]
//   Fragment components selected branchlessly (v_cndmask, no exec churn, EXEC
//   stays all-ones around the WMMA as required by ISA 7.12).
// ---------------------------------------------------------------------------
#define CHUNK 512

__global__ void knn_kernel(const float* __restrict__ xyz,
                           const float* __restrict__ new_xyz_ws,
                           int* __restrict__ knn_idx) {
  const int tile = blockIdx.x;                 // B * (S/16) tiles
  const int b = tile / (SS / 16);
  const int s0 = (tile % (SS / 16)) * 16;
  const float* X = xyz + (size_t)b * NN * 3;
  const float* Q = new_xyz_ws + ((size_t)b * SS + s0) * 3;

  __shared__ float s_dist[16 * CHUNK];         // 32 KB
  __shared__ float s_q[16 * 3];

  const int tid = threadIdx.x;                 // 256 threads = 8 waves
  const int lane = tid & 31;
  const int wave = tid >> 5;
  const bool lo = (lane < 16);
  if (tid < 48) s_q[tid] = Q[tid];
  __syncthreads();

  // A fragment (16x4 f32): lanes 0-15 -> {K0,K1}, lanes 16-31 -> {K2,K3}
  const int m = lane & 15;
  const float qx = s_q[m * 3 + 0], qy = s_q[m * 3 + 1], qz = s_q[m * 3 + 2];
  v2f afrag;
  afrag.x = lo ? (-2.0f * qx) : (-2.0f * qz);
  afrag.y = lo ? (-2.0f * qy) : 1.0f;

  // top-24 running state (threads 0..15, one per query row)
  float bd[KK]; int bi[KK]; float worst = 1e30f; int wpos = 0;
  if (tid < 16)
    for (int k = 0; k < KK; ++k) { bd[k] = 1e30f; bi[k] = 0; }

  for (int base = 0; base < NN; base += CHUNK) {
    for (int t = wave; t < CHUNK / 16; t += 8) {     // wave-uniform trip count
      const int pi = base + t * 16 + (lane & 15);
      const float px = X[3 * pi + 0], py = X[3 * pi + 1], pz = X[3 * pi + 2];
      const float psq = px * px + py * py + pz * pz;
      v2f bfrag;
      bfrag.x = lo ? px : pz;       // branchless: v_cndmask, no exec dance
      bfrag.y = lo ? py : psq;
      v8f c = {};
      c = __builtin_amdgcn_wmma_f32_16x16x4_f32(
          /*neg_a=*/false, afrag, /*neg_b=*/false, bfrag,
          /*c_mod=*/(short)0, c, /*reuse_a=*/false, /*reuse_b=*/false);
      const int col = t * 16 + (lane & 15);
      const int rbase = lo ? 0 : 8;
#pragma unroll
      for (int r = 0; r < 8; ++r)
        s_dist[(rbase + r) * CHUNK + col] = c[r];
    }
    __syncthreads();
    if (tid < 16) {
      // row base is 2KB aligned -> ds_load_b128 scan
      const float4* row = (const float4*)&s_dist[tid * CHUNK];
      for (int j4 = 0; j4 < CHUNK / 4; ++j4) {
        const float4 d4 = row[j4];
        const float dv[4] = {d4.x, d4.y, d4.z, d4.w};
#pragma unroll
        for (int u = 0; u < 4; ++u) {
          const float d = dv[u];
          if (d < worst) {
            bd[wpos] = d; bi[wpos] = base + j4 * 4 + u;
            worst = bd[0]; wpos = 0;
            for (int k = 1; k < KK; ++k)
              if (bd[k] > worst) { worst = bd[k]; wpos = k; }
          }
        }
      }
    }
    __syncthreads();
  }

  if (tid < 16) {
    // sort the 24 picks ascending by distance (index tiebreak)
    for (int a = 0; a < KK - 1; ++a) {
      int mbest = a;
      for (int c2 = a + 1; c2 < KK; ++c2)
        if (bd[c2] < bd[mbest] || (bd[c2] == bd[mbest] && bi[c2] < bi[mbest])) mbest = c2;
      float td = bd[a]; bd[a] = bd[mbest]; bd[mbest] = td;
      int ti = bi[a]; bi[a] = bi[mbest]; bi[mbest] = ti;
    }
    const size_t o = ((size_t)b * SS + (s0 + tid)) * KK;
    for (int k = 0; k < KK; ++k) knn_idx[o + k] = bi[k];
  }
}

// ---------------------------------------------------------------------------
// Kernel 4: per-(b,s) channel means + per-batch sum of squared centered values
// ---------------------------------------------------------------------------
__global__ void zero_kernel(float* __restrict__ p, int n) {
  int i = blockIdx.x * blockDim.x + threadIdx.x;
  if (i < n) p[i] = 0.0f;
}

__global__ void stats_kernel(const float* __restrict__ xyz,
                             const float* __restrict__ points,
                             const int* __restrict__ knn_idx,
                             float* __restrict__ mean_ws,
                             float* __restrict__ batch_sumsq) {
  const int bs = blockIdx.x;        // B*S blocks
  const int b = bs / SS;
  const int* nb = knn_idx + (size_t)bs * KK;
  float partial = 0.0f;
  for (int c = threadIdx.x; c < CIN; c += blockDim.x) {
    float sum = 0.0f, sumsq = 0.0f;
#pragma unroll 4
    for (int k = 0; k < KK; ++k) {
      const int id = nb[k];
      const float v = (c < DD) ? points[((size_t)b * NN + id) * DD + c]
                               : xyz[((size_t)b * NN + id) * 3 + (c - DD)];
      sum += v; sumsq += v * v;
    }
    const float mean = sum * (1.0f / KK);
    mean_ws[(size_t)bs * CIN + c] = mean;
    partial += sumsq - (float)KK * mean * mean;  // sum_k (x-mean)^2
  }
  __shared__ float red[256];
  red[threadIdx.x] = partial;
  __syncthreads();
  for (int off = 128; off > 0; off >>= 1) {
    if (threadIdx.x < off) red[threadIdx.x] += red[threadIdx.x + off];
    __syncthreads();
  }
  if (threadIdx.x == 0) atomicAdd(&batch_sumsq[b], red[0]);
}

// ---------------------------------------------------------------------------
// Kernel 5: per-batch reciprocal std (ddof = 1)
// ---------------------------------------------------------------------------
__global__ void rstd_kernel(const float* __restrict__ batch_sumsq,
                            float* __restrict__ batch_rstd) {
  int t = threadIdx.x;
  if (t < BB) {
    const double count = (double)SS * KK * CIN;
    float var = (float)((double)batch_sumsq[t] / (count - 1.0));
    batch_rstd[t] = 1.0f / (sqrtf(var) + EPS);
  }
}

// ---------------------------------------------------------------------------
// Kernel 6: normalized grouped features + anchor concat -> new_points_out
// ---------------------------------------------------------------------------
__global__ void final_kernel(const float* __restrict__ xyz,
                             const float* __restrict__ points,
                             const int* __restrict__ knn_idx,
                             const int* __restrict__ fps_idx,
                             const float* __restrict__ mean_ws,
                             const float* __restrict__ batch_rstd,
                             const float* __restrict__ alpha,
                             const float* __restrict__ beta,
                             float* __restrict__ out2) {
  const int g = blockIdx.x;              // B*S*K blocks
  const int k = g % KK;
  const int bs = g / KK;
  const int b = bs / SS;
  const int id = knn_idx[(size_t)bs * KK + k];
  const int anchor = fps_idx[bs];
  const float r = batch_rstd[b];
  const float* mrow = mean_ws + (size_t)bs * CIN;
  const float* arow = points + ((size_t)b * NN + anchor) * DD;
  float* o = out2 + (size_t)g * COUT;

  // Cover anchor-row latency: the anchor loads happen on the last loop
  // iterations; pull its 8 cachelines (1 KB) toward the WGP now.
  // Lowers to global_prefetch_b8 on gfx1250.
  if (threadIdx.x < 8)
    __builtin_prefetch(arow + threadIdx.x * 32, 0, 0);

  for (int c = threadIdx.x; c < COUT; c += blockDim.x) {
    float v;
    if (c < CIN) {
      const float gv = (c < DD) ? points[((size_t)b * NN + id) * DD + c]
                                : xyz[((size_t)b * NN + id) * 3 + (c - DD)];
      v = alpha[c] * ((gv - mrow[c]) * r) + beta[c];
    } else {
      v = arow[c - CIN];
    }
    o[c] = v;
  }
}

// ---------------------------------------------------------------------------
// Launch
// ---------------------------------------------------------------------------
extern "C" void kernel_launch(void* const* d_in, const int* in_sizes, int n_in,
                              void* d_out, int out_size, void* d_ws, size_t ws_size,
                              hipStream_t stream) {
  const float* xyz        = (const float*)d_in[0];   // [B,N,3]
  const float* points     = (const float*)d_in[1];   // [B,N,D]
  const float* points_res = (const float*)d_in[2];   // [B,N,D]
  const float* alpha      = (const float*)d_in[3];   // [D+3]
  const float* beta       = (const float*)d_in[4];   // [D+3]

  // Output layout (flat, return order): new_xyz | new_points_out | new_points_res
  float* out_xyz = (float*)d_out;                                   // B*S*3
  float* out2    = out_xyz + (size_t)BB * SS * 3;                   // B*S*K*(2D+3)
  float* out_res = out2 + (size_t)BB * SS * KK * COUT;              // B*S*D

  // Workspace layout
  char* ws = (char*)d_ws;
  size_t off = 0;
  auto take = [&](size_t bytes) { void* p = ws + off; off = (off + bytes + 255) & ~(size_t)255; return p; };
  int*   fps_idx     = (int*)  take((size_t)BB * SS * sizeof(int));
  int*   knn_idx     = (int*)  take((size_t)BB * SS * KK * sizeof(int));
  float* new_xyz_ws  = (float*)take((size_t)BB * SS * 3 * sizeof(float));
  float* mean_ws     = (float*)take((size_t)BB * SS * CIN * sizeof(float));
  float* batch_sumsq = (float*)take((size_t)BB * sizeof(float));
  float* batch_rstd  = (float*)take((size_t)BB * sizeof(float));

  zero_kernel<<<1, 64, 0, stream>>>(batch_sumsq, BB);
  fps_kernel<<<BB, 256, 0, stream>>>(xyz, fps_idx);
  gather_kernel<<<BB * SS, 256, 0, stream>>>(xyz, points_res, fps_idx,
                                             out_xyz, new_xyz_ws, out_res);
  knn_kernel<<<BB * (SS / 16), 256, 0, stream>>>(xyz, new_xyz_ws, knn_idx);
  stats_kernel<<<BB * SS, 256, 0, stream>>>(xyz, points, knn_idx, mean_ws, batch_sumsq);
  rstd_kernel<<<1, 64, 0, stream>>>(batch_sumsq, batch_rstd);
  final_kernel<<<BB * SS * KK, 256, 0, stream>>>(xyz, points, knn_idx, fps_idx,
                                                 mean_ws, batch_rstd, alpha, beta, out2);
}